// GINENet_59090160059185
// MI455X (gfx1250) — compile-verified
//
#include <hip/hip_runtime.h>
#include <hip/hip_bf16.h>

#define N_NODES  100000
#define N_EDGES  1600000
#define N_GRAPHS 256
#define F_NODE   64
#define F_EDGE   16
#define HIDDEN   64
#define N_TASKS  10

typedef float v2f __attribute__((ext_vector_type(2)));
typedef float v8f __attribute__((ext_vector_type(8)));

// ---------------------------------------------------------------- utilities
__global__ void zero_f32(float* __restrict__ p, int n) {
    for (int i = blockIdx.x * blockDim.x + threadIdx.x; i < n;
         i += gridDim.x * blockDim.x)
        p[i] = 0.0f;
}

// Fold the two edge linears:  Wc = We1 @ We2  (16x64),  bc = be1 @ We2 + be2
__global__ void fuse_edge_weights(const float* __restrict__ We1,
                                  const float* __restrict__ be1,
                                  const float* __restrict__ We2,
                                  const float* __restrict__ be2,
                                  float* __restrict__ Wc,
                                  float* __restrict__ bc) {
    int f = threadIdx.x;                 // 0..63
    float accb = be2[f];
    for (int j = 0; j < 64; ++j) accb += be1[j] * We2[j * 64 + f];
    bc[f] = accb;
    for (int k = 0; k < 16; ++k) {
        float acc = 0.0f;
        for (int j = 0; j < 64; ++j) acc += We1[k * 64 + j] * We2[j * 64 + f];
        Wc[k * 64 + f] = acc;
    }
}

// ------------------------------------------------- scatter (message) pass
// 16 threads per edge, 4 features each. Edge feature (16 -> 64 linear) is
// recomputed on the fly; msg = relu(h[src] + ea); atomic add into agg[dst].
__global__ void scatter_edges(const float* __restrict__ h,
                              const int*   __restrict__ edge_index, // [2, E]
                              const float* __restrict__ edge_attr,  // [E, 16]
                              const float* __restrict__ We,         // [16, 64]
                              const float* __restrict__ be,         // [64]
                              float*       __restrict__ agg) {      // [N, 64]
    const int total = N_EDGES * 16;
    for (int t = blockIdx.x * blockDim.x + threadIdx.x; t < total;
         t += gridDim.x * blockDim.x) {
        int e   = t >> 4;
        int f0  = (t & 15) * 4;
        int src = edge_index[e];
        int dst = edge_index[N_EDGES + e];

        float4 acc = *(const float4*)(be + f0);
        const float* ea = edge_attr + e * 16;
#pragma unroll
        for (int k = 0; k < 16; ++k) {
            float a  = ea[k];
            float4 w = *(const float4*)(We + k * 64 + f0);
            acc.x += a * w.x; acc.y += a * w.y;
            acc.z += a * w.z; acc.w += a * w.w;
        }
        float4 hv = *(const float4*)(h + src * 64 + f0);
        float mx = fmaxf(hv.x + acc.x, 0.0f);
        float my = fmaxf(hv.y + acc.y, 0.0f);
        float mz = fmaxf(hv.z + acc.z, 0.0f);
        float mw = fmaxf(hv.w + acc.w, 0.0f);
        float* dp = agg + dst * 64 + f0;
        unsafeAtomicAdd(dp + 0, mx);
        unsafeAtomicAdd(dp + 1, my);
        unsafeAtomicAdd(dp + 2, mz);
        unsafeAtomicAdd(dp + 3, mw);
    }
}

// ------------------------------------------- WMMA node update GEMM (64x64)
// out = maybe_relu( ((1+eps)*h + agg) @ W + b )
// One wave32 computes a 16-row x 64-col tile with V_WMMA_F32_16X16X4_F32.
// If POOL: fused scatter-mean numerator -> atomicAdd into pooled[batch[row]].
template <bool RELU, bool POOL>
__global__ void node_update(const float* __restrict__ h,
                            const float* __restrict__ agg,
                            const float* __restrict__ W,     // [64, 64]
                            const float* __restrict__ bias,  // [64]
                            const float* __restrict__ epsPtr,
                            const int*   __restrict__ batch,
                            float*       __restrict__ out) { // h_out or pooled
    const int NT   = N_NODES / 16;                     // 6250 row-tiles
    int wave = blockIdx.x * (blockDim.x >> 5) + (threadIdx.x >> 5);
    if (wave >= NT) return;                            // wave-uniform: EXEC full

    int lane = threadIdx.x & 31;
    int hi   = lane >> 4;                              // half-wave select
    int m    = lane & 15;
    int rowA = wave * 16 + m;
    float scale = 1.0f + epsPtr[0];

    v8f c0 = {}, c1 = {}, c2 = {}, c3 = {};
#pragma unroll
    for (int k0 = 0; k0 < 64; k0 += 4) {
        int k = k0 + 2 * hi;                           // ISA 16x4 f32 A layout
        v2f hv = *(const v2f*)(h   + rowA * 64 + k);
        v2f av = *(const v2f*)(agg + rowA * 64 + k);
        v2f a;
        a.x = scale * hv.x + av.x;
        a.y = scale * hv.y + av.y;
        const float* Wk0 = W + k * 64;
        const float* Wk1 = W + (k + 1) * 64;
        v2f b;
        b.x = Wk0[m];        b.y = Wk1[m];
        c0 = __builtin_amdgcn_wmma_f32_16x16x4_f32(false, a, false, b, (short)0, c0, false, false);
        b.x = Wk0[16 + m];   b.y = Wk1[16 + m];
        c1 = __builtin_amdgcn_wmma_f32_16x16x4_f32(false, a, false, b, (short)0, c1, false, false);
        b.x = Wk0[32 + m];   b.y = Wk1[32 + m];
        c2 = __builtin_amdgcn_wmma_f32_16x16x4_f32(false, a, false, b, (short)0, c2, false, false);
        b.x = Wk0[48 + m];   b.y = Wk1[48 + m];
        c3 = __builtin_amdgcn_wmma_f32_16x16x4_f32(false, a, false, b, (short)0, c3, false, false);
    }

    v8f cc[4] = {c0, c1, c2, c3};
#pragma unroll
    for (int nt = 0; nt < 4; ++nt) {
#pragma unroll
        for (int v = 0; v < 8; ++v) {                  // C/D layout: M=v+8*hi, N=m
            int row = wave * 16 + v + 8 * hi;
            int col = nt * 16 + m;
            float val = cc[nt][v] + bias[col];
            if (RELU) val = fmaxf(val, 0.0f);
            if (POOL) {
                int g = batch[row];
                unsafeAtomicAdd(out + g * 64 + col, val);
            } else {
                out[row * 64 + col] = val;
            }
        }
    }
}

__global__ void count_nodes(const int* __restrict__ batch,
                            float* __restrict__ counts) {
    int i = blockIdx.x * blockDim.x + threadIdx.x;
    if (i < N_NODES) unsafeAtomicAdd(&counts[batch[i]], 1.0f);
}

// -------------------------------------------------------------- MLP head
__global__ void head_mlp(const float* __restrict__ pooled,
                         const float* __restrict__ counts,
                         const float* __restrict__ Wm1,
                         const float* __restrict__ bm1,
                         const float* __restrict__ Wm2,
                         const float* __restrict__ bm2,
                         float* __restrict__ out) {
    int g = blockIdx.x * blockDim.x + threadIdx.x;
    if (g >= N_GRAPHS) return;
    float inv = 1.0f / fmaxf(counts[g], 1.0f);
    float p[64];
#pragma unroll
    for (int k = 0; k < 64; ++k) p[k] = pooled[g * 64 + k] * inv;
    float o[N_TASKS];
#pragma unroll
    for (int t = 0; t < N_TASKS; ++t) o[t] = bm2[t];
    for (int j = 0; j < 64; ++j) {
        float hsum = bm1[j];
#pragma unroll
        for (int k = 0; k < 64; ++k) hsum += p[k] * Wm1[k * 64 + j];
        hsum = fmaxf(hsum, 0.0f);
#pragma unroll
        for (int t = 0; t < N_TASKS; ++t) o[t] += hsum * Wm2[j * N_TASKS + t];
    }
#pragma unroll
    for (int t = 0; t < N_TASKS; ++t) out[g * N_TASKS + t] = o[t];
}

// ---------------------------------------------------------------- launch
extern "C" void kernel_launch(void* const* d_in, const int* in_sizes, int n_in,
                              void* d_out, int out_size, void* d_ws, size_t ws_size,
                              hipStream_t stream) {
    const float* x          = (const float*)d_in[0];
    const int*   edge_index = (const int*)d_in[1];
    const float* edge_attr  = (const float*)d_in[2];
    const int*   batch      = (const int*)d_in[3];
    const float* We1 = (const float*)d_in[4];
    const float* be1 = (const float*)d_in[5];
    const float* We2 = (const float*)d_in[6];
    const float* be2 = (const float*)d_in[7];
    const float* W1  = (const float*)d_in[8];
    const float* b1  = (const float*)d_in[9];
    const float* eps1 = (const float*)d_in[10];
    const float* W2  = (const float*)d_in[11];
    const float* b2  = (const float*)d_in[12];
    const float* eps2 = (const float*)d_in[13];
    const float* W3  = (const float*)d_in[14];
    const float* b3  = (const float*)d_in[15];
    const float* eps3 = (const float*)d_in[16];
    const float* Wm1 = (const float*)d_in[17];
    const float* bm1 = (const float*)d_in[18];
    const float* Wm2 = (const float*)d_in[19];
    const float* bm2 = (const float*)d_in[20];
    float* out = (float*)d_out;

    // Workspace carve-up (floats):
    float* w      = (float*)d_ws;
    float* Wc     = w;                         // 16*64
    float* bc     = Wc + 16 * 64;              // 64
    float* agg    = bc + 64;                   // N*64
    float* hA     = agg + N_NODES * 64;        // N*64
    float* hB     = hA + N_NODES * 64;         // N*64
    float* pooled = hB + N_NODES * 64;         // G*64
    float* counts = pooled + N_GRAPHS * 64;    // G

    const int aggN      = N_NODES * 64;
    const int zeroBlks  = 4096;
    const int scatBlks  = 8192;                // grid-stride over E*16 threads
    const int nodeTiles = N_NODES / 16;        // 6250
    const int nodeBlks  = (nodeTiles + 3) / 4; // 4 waves / 128-thread block

    fuse_edge_weights<<<1, 64, 0, stream>>>(We1, be1, We2, be2, Wc, bc);

    // ---- conv1: edge weights = We1/be1, node input = x
    zero_f32<<<zeroBlks, 256, 0, stream>>>(agg, aggN);
    scatter_edges<<<scatBlks, 256, 0, stream>>>(x, edge_index, edge_attr, We1, be1, agg);
    node_update<true, false><<<nodeBlks, 128, 0, stream>>>(x, agg, W1, b1, eps1, nullptr, hA);

    // ---- conv2: edge weights = fused Wc/bc, node input = hA
    zero_f32<<<zeroBlks, 256, 0, stream>>>(agg, aggN);
    scatter_edges<<<scatBlks, 256, 0, stream>>>(hA, edge_index, edge_attr, Wc, bc, agg);
    node_update<true, false><<<nodeBlks, 128, 0, stream>>>(hA, agg, W2, b2, eps2, nullptr, hB);

    // ---- conv3 (no relu) fused with graph pooling
    zero_f32<<<zeroBlks, 256, 0, stream>>>(agg, aggN);
    zero_f32<<<64, 256, 0, stream>>>(pooled, N_GRAPHS * 64);
    zero_f32<<<1, 256, 0, stream>>>(counts, N_GRAPHS);
    scatter_edges<<<scatBlks, 256, 0, stream>>>(hB, edge_index, edge_attr, Wc, bc, agg);
    count_nodes<<<(N_NODES + 255) / 256, 256, 0, stream>>>(batch, counts);
    node_update<false, true><<<nodeBlks, 128, 0, stream>>>(hB, agg, W3, b3, eps3, batch, pooled);

    // ---- head
    head_mlp<<<(N_GRAPHS + 63) / 64, 64, 0, stream>>>(pooled, counts, Wm1, bm1, Wm2, bm2, out);
}